// Model_53214644797709
// MI455X (gfx1250) — compile-verified
//
#include <hip/hip_runtime.h>

// Problem constants (from reference): B=16, S=8, D=512, H=2048, 10 steps, lr=0.01
constexpr int BB = 16;
constexpr int SS = 8;
constexpr int DD = 512;
constexpr int HH = 2048;
constexpr int NSTEP = 10;
#define LRI 0.01f

typedef __attribute__((ext_vector_type(2))) float v2f;
typedef __attribute__((ext_vector_type(8))) float v8f;

// ---------------------------------------------------------------------------
// WMMA f32 16x16x4 wrapper. 8-arg form per probe: (neg_a, A, neg_b, B, c_mod,
// C, reuse_a, reuse_b). A = 16x4 f32 (2 VGPR), B = 4x16 f32 (2 VGPR), C/D =
// 16x16 f32 (8 VGPR).
// ---------------------------------------------------------------------------
__device__ __forceinline__ v8f wmma4(v2f a, v2f b, v8f c) {
  return __builtin_amdgcn_wmma_f32_16x16x4_f32(
      /*neg_a=*/false, a, /*neg_b=*/false, b,
      /*c_mod=*/(short)0, c, /*reuse_a=*/false, /*reuse_b=*/false);
}

// A-matrix 16x4 f32 layout (ISA 7.12.2): lanes 0-15 hold M=lane with K=0(.x),
// K=1(.y); lanes 16-31 hold M=lane-16 with K=2(.x), K=3(.y).
__device__ __forceinline__ v2f load_a(const float* A, int lda, int k0) {
  int lane = threadIdx.x & 31;
  int m = lane & 15;
  int kk = (lane >> 4) << 1;  // 0 or 2
  v2f a;
  a.x = A[m * lda + k0 + kk];
  a.y = A[m * lda + k0 + kk + 1];
  return a;
}

// B-matrix 4x16 f32, row-major source: B[k][n] = Bp[k*ldb + n0 + n].
// Mirrors the A layout: lanes 0-15 -> N=lane, rows K=0(.x),1(.y);
// lanes 16-31 -> N=lane-16, rows K=2(.x),3(.y).
__device__ __forceinline__ v2f load_b_rm(const float* Bp, int ldb, int k0, int n0) {
  int lane = threadIdx.x & 31;
  int n = lane & 15;
  int kk = (lane >> 4) << 1;
  v2f b;
  b.x = Bp[(k0 + kk) * ldb + n0 + n];
  b.y = Bp[(k0 + kk + 1) * ldb + n0 + n];
  return b;
}

// B = Mᵀ with M row-major: B[k][n] = Mp[(n0+n)*ldm + k].
__device__ __forceinline__ v2f load_b_tr(const float* Mp, int ldm, int k0, int n0) {
  int lane = threadIdx.x & 31;
  int n = lane & 15;
  int kk = (lane >> 4) << 1;
  v2f b;
  b.x = Mp[(n0 + n) * ldm + k0 + kk];
  b.y = Mp[(n0 + n) * ldm + k0 + kk + 1];
  return b;
}

// C/D 16x16 f32 layout (documented): lanes 0-15: M=r, N=lane;
// lanes 16-31: M=r+8, N=lane-16.
__device__ __forceinline__ void cpos(int r, int& m, int& n) {
  int lane = threadIdx.x & 31;
  n = lane & 15;
  m = r + ((lane >> 4) << 3);
}

// ---------------------------------------------------------------------------
// Utility kernels
// ---------------------------------------------------------------------------
__global__ void zero_buf(float* p, int n) {
  int i = blockIdx.x * blockDim.x + threadIdx.x;
  if (i < n) p[i] = 0.0f;
}

// c[b] = ii·ii, e[b] = oo·ii   (ii = inp[:, -2], oo = inp[:, -1])
__global__ void dots_ii(const float* inp, float* cbuf, float* ebuf) {
  int b = blockIdx.x;
  const float* ii = inp + (size_t)b * SS * DD + 6 * DD;
  const float* oo = inp + (size_t)b * SS * DD + 7 * DD;
  __shared__ float red[256];
  __shared__ float red2[256];
  float cc = 0.f, ee = 0.f;
  for (int d = threadIdx.x; d < DD; d += 256) {
    float v = ii[d];
    cc += v * v;
    ee += oo[d] * v;
  }
  red[threadIdx.x] = cc;
  red2[threadIdx.x] = ee;
  __syncthreads();
  for (int off = 128; off > 0; off >>= 1) {
    if ((int)threadIdx.x < off) {
      red[threadIdx.x] += red[threadIdx.x + off];
      red2[threadIdx.x] += red2[threadIdx.x + off];
    }
    __syncthreads();
  }
  if (threadIdx.x == 0) { cbuf[b] = red[0]; ebuf[b] = red2[0]; }
}

// A0 = II @ l1, AF = OO @ l1   (both 16 x 2048, K = 512). One wave per 16x16
// output tile; grid = 2 * (2048/16) = 256.
__global__ void prep_gemm(const float* inp, const float* l1, float* A0, float* AF) {
  int id = blockIdx.x;
  int which = id >> 7;             // 0 -> II (row -2), 1 -> OO (row -1)
  int n0 = (id & 127) << 4;
  const float* A = inp + (which ? 7 * DD : 6 * DD);  // row stride = S*D
  float* out = which ? AF : A0;
  v8f c = {};
  for (int k0 = 0; k0 < DD; k0 += 4) {
    v2f a = load_a(A, SS * DD, k0);
    v2f b = load_b_rm(l1, HH, k0, n0);
    c = wmma4(a, b, c);
  }
  for (int r = 0; r < 8; ++r) {
    int m, n; cpos(r, m, n);
    out[m * HH + n0 + n] = c[r];
  }
}

// h1 = relu(pre - scal*u); store into Hhist[t]; hdots[b][s] = h1 · Hhist[s]
// (s < t). Grid = 16 (one block per batch), block = 256 (8 elems/thread).
// Also used for the final forward pass with pre = AF, scal = e, t = NSTEP.
__global__ void h1_kernel(const float* pre, const float* scal, const float* u,
                          float* Hhist, float* dots, int t) {
  int b = blockIdx.x;
  float cc = scal[b];
  float h1v[8];
  int idx[8];
  float* hcur = Hhist + (size_t)t * BB * HH + (size_t)b * HH;
  const float* prow = pre + (size_t)b * HH;
  const float* urow = u + (size_t)b * HH;
  for (int i = 0; i < 8; ++i) {
    int h = threadIdx.x + i * 256;
    idx[i] = h;
    float p = prow[h] - cc * urow[h];
    float v = p > 0.f ? p : 0.f;
    h1v[i] = v;
    hcur[h] = v;
  }
  __shared__ float red[256];
  for (int s = 0; s < t; ++s) {
    const float* hs = Hhist + (size_t)s * BB * HH + (size_t)b * HH;
    float p = 0.f;
    for (int i = 0; i < 8; ++i) p += h1v[i] * hs[idx[i]];
    red[threadIdx.x] = p;
    __syncthreads();
    for (int off = 128; off > 0; off >>= 1) {
      if ((int)threadIdx.x < off) red[threadIdx.x] += red[threadIdx.x + off];
      __syncthreads();
    }
    if (threadIdx.x == 0) dots[b * 16 + s] = red[0];
    __syncthreads();
  }
}

// pred_raw = Hhist[t] @ l2 (16x512, K=2048); fuse rank corrections and the
// dpred conversion: dpred = 2*(pred - oo)/(B*D), stored in Dhist[t].
// Grid = 32 waves (one 16x16 tile each), block = 32.
__global__ void pred_kernel(const float* Hhist, const float* l2, const float* inp,
                            const float* dots, float* Dhist, int t) {
  int n0 = blockIdx.x << 4;
  const float* A = Hhist + (size_t)t * BB * HH;
  v8f c = {};
  for (int k0 = 0; k0 < HH; k0 += 4) {
    v2f a = load_a(A, HH, k0);
    v2f b = load_b_rm(l2, DD, k0, n0);
    c = wmma4(a, b, c);
  }
  float* dcur = Dhist + (size_t)t * BB * DD;
  const float scale = 2.0f / (float)(BB * DD);
  for (int r = 0; r < 8; ++r) {
    int m, n; cpos(r, m, n);
    int nn = n0 + n;
    float val = c[r];
    for (int s = 0; s < t; ++s)
      val -= LRI * dots[m * 16 + s] * Dhist[(size_t)s * BB * DD + m * DD + nn];
    float oo = inp[(size_t)m * SS * DD + 7 * DD + nn];
    dcur[m * DD + nn] = (val - oo) * scale;
  }
}

// ddots[b][s] = Dhist[t][b] · Dhist[s][b]  (s < t). Grid = 16, block = 256.
__global__ void ddots_kernel(const float* Dhist, float* dots, int t) {
  int b = blockIdx.x;
  const float* dt = Dhist + (size_t)t * BB * DD + (size_t)b * DD;
  __shared__ float red[256];
  float v0 = dt[threadIdx.x];
  float v1 = dt[threadIdx.x + 256];
  for (int s = 0; s < t; ++s) {
    const float* ds = Dhist + (size_t)s * BB * DD + (size_t)b * DD;
    float p = v0 * ds[threadIdx.x] + v1 * ds[threadIdx.x + 256];
    red[threadIdx.x] = p;
    __syncthreads();
    for (int off = 128; off > 0; off >>= 1) {
      if ((int)threadIdx.x < off) red[threadIdx.x] += red[threadIdx.x + off];
      __syncthreads();
    }
    if (threadIdx.x == 0) dots[b * 16 + s] = red[0];
    __syncthreads();
  }
}

// dh_raw = Dhist[t] @ l2ᵀ (16x2048, K=512); fuse rank corrections, the ReLU
// mask (h1 > 0), and the SGD accumulate u += lr*dh. Grid = 128 waves.
__global__ void dh_kernel(const float* Dhist, const float* l2, const float* Hhist,
                          const float* dots, float* u, int t) {
  int n0 = blockIdx.x << 4;  // over H
  const float* A = Dhist + (size_t)t * BB * DD;
  v8f c = {};
  for (int k0 = 0; k0 < DD; k0 += 4) {
    v2f a = load_a(A, DD, k0);
    v2f b = load_b_tr(l2, DD, k0, n0);  // B[k][n] = l2[(n0+n)*512 + k]
    c = wmma4(a, b, c);
  }
  const float* hcur = Hhist + (size_t)t * BB * HH;
  for (int r = 0; r < 8; ++r) {
    int m, n; cpos(r, m, n);
    int hh = n0 + n;
    float val = c[r];
    for (int s = 0; s < t; ++s)
      val -= LRI * dots[m * 16 + s] * Hhist[(size_t)s * BB * HH + m * HH + hh];
    if (hcur[m * HH + hh] > 0.f)
      u[m * HH + hh] += LRI * val;
  }
}

// Final: pred = Hhist[NSTEP] @ l2 - lr * Σ_s fdots[s] * Dhist[s] -> d_out
__global__ void predf_kernel(const float* Hhist, const float* l2,
                             const float* dots, const float* Dhist, float* outp) {
  int n0 = blockIdx.x << 4;
  const float* A = Hhist + (size_t)NSTEP * BB * HH;
  v8f c = {};
  for (int k0 = 0; k0 < HH; k0 += 4) {
    v2f a = load_a(A, HH, k0);
    v2f b = load_b_rm(l2, DD, k0, n0);
    c = wmma4(a, b, c);
  }
  for (int r = 0; r < 8; ++r) {
    int m, n; cpos(r, m, n);
    int nn = n0 + n;
    float val = c[r];
    for (int s = 0; s < NSTEP; ++s)
      val -= LRI * dots[m * 16 + s] * Dhist[(size_t)s * BB * DD + m * DD + nn];
    outp[m * DD + nn] = val;
  }
}

// ---------------------------------------------------------------------------
extern "C" void kernel_launch(void* const* d_in, const int* in_sizes, int n_in,
                              void* d_out, int out_size, void* d_ws, size_t ws_size,
                              hipStream_t stream) {
  (void)in_sizes; (void)n_in; (void)out_size; (void)ws_size;
  const float* inp = (const float*)d_in[0];  // (16, 8, 512)
  const float* l1  = (const float*)d_in[1];  // (512, 2048)
  const float* l2  = (const float*)d_in[2];  // (2048, 512)
  float* out = (float*)d_out;                // (16, 512)
  float* ws = (float*)d_ws;

  // Workspace layout (floats): ~2.1 MB total
  float* A0 = ws;                                   // 16*2048
  float* AF = A0 + BB * HH;                         // 16*2048
  float* u  = AF + BB * HH;                         // 16*2048
  float* Hh = u + BB * HH;                          // (NSTEP+1)*16*2048
  float* Dh = Hh + (size_t)(NSTEP + 1) * BB * HH;   // NSTEP*16*512
  float* cb = Dh + (size_t)NSTEP * BB * DD;         // 16
  float* eb = cb + BB;                              // 16
  float* dots = eb + BB;                            // 16*16

  zero_buf<<<(BB * HH + 255) / 256, 256, 0, stream>>>(u, BB * HH);
  dots_ii<<<BB, 256, 0, stream>>>(inp, cb, eb);
  prep_gemm<<<256, 32, 0, stream>>>(inp, l1, A0, AF);

  for (int t = 0; t < NSTEP; ++t) {
    h1_kernel<<<BB, 256, 0, stream>>>(A0, cb, u, Hh, dots, t);
    pred_kernel<<<DD / 16, 32, 0, stream>>>(Hh, l2, inp, dots, Dh, t);
    ddots_kernel<<<BB, 256, 0, stream>>>(Dh, dots, t);
    dh_kernel<<<HH / 16, 32, 0, stream>>>(Dh, l2, Hh, dots, u, t);
  }

  // Final forward pass with x = oo
  h1_kernel<<<BB, 256, 0, stream>>>(AF, eb, u, Hh, dots, NSTEP);
  predf_kernel<<<DD / 16, 32, 0, stream>>>(Hh, l2, dots, Dh, out);
}